// GraphMultiHeadAttentionEinSum_47175920779378
// MI455X (gfx1250) — compile-verified
//
#include <hip/hip_runtime.h>
#include <hip/hip_bf16.h>
#include <math.h>

// Problem dims (fixed by the reference)
static constexpr int Bc = 4;
static constexpr int Nn = 1024;
static constexpr int Dd = 256;
static constexpr int Hh = 8;
static constexpr int DH = 32;   // Dd / Hh

typedef float v2f __attribute__((ext_vector_type(2)));
typedef float v8f __attribute__((ext_vector_type(8)));

// ---------------------------------------------------------------------------
// Kernel 1: Wh1 = h @ W1^T, Wh2 = h @ W2^T   (fp32 WMMA 16x16x4)
// One wave per 16x16 output tile; shared A-tile feeds two WMMA chains.
// A layout: lane(0..15) M=lane K={0,1}; lane(16..31) M=lane-16 K={2,3}
// B layout: lane(0..15) N=lane K={0,1}; lane(16..31) N=lane-16 K={2,3}
// ---------------------------------------------------------------------------
__global__ __launch_bounds__(32) void proj12_kernel(
    const float* __restrict__ hmat, const float* __restrict__ W1,
    const float* __restrict__ W2, float* __restrict__ wh1,
    float* __restrict__ wh2)
{
    const int lane = threadIdx.x & 31;
    const int m    = lane & 15;
    const int kh   = (lane >> 4) << 1;        // 0 or 2
    const int col0 = blockIdx.x * 16;
    const int row0 = blockIdx.y * 16;

    v8f acc1 = {};
    v8f acc2 = {};
    const float* aRow  = hmat + (long)(row0 + m) * Dd;
    const float* b1Row = W1   + (long)(col0 + m) * Dd;   // out[r,c]=sum_k h[r,k]*W[c,k]
    const float* b2Row = W2   + (long)(col0 + m) * Dd;

    for (int k0 = 0; k0 < Dd; k0 += 4) {
        v2f a  = *(const v2f*)(aRow  + k0 + kh);
        v2f b1 = *(const v2f*)(b1Row + k0 + kh);
        v2f b2 = *(const v2f*)(b2Row + k0 + kh);
        acc1 = __builtin_amdgcn_wmma_f32_16x16x4_f32(false, a, false, b1,
                                                     (short)0, acc1, false, false);
        acc2 = __builtin_amdgcn_wmma_f32_16x16x4_f32(false, a, false, b2,
                                                     (short)0, acc2, false, false);
    }

    // C/D layout: VGPR r, lanes 0-15 -> M=r, lanes 16-31 -> M=r+8; N=lane&15
    const int rbase = row0 + ((lane >> 4) << 3);
    const int c     = col0 + m;
#pragma unroll
    for (int r = 0; r < 8; ++r) {
        wh1[(long)(rbase + r) * Dd + c] = acc1[r];
        wh2[(long)(rbase + r) * Dd + c] = acc2[r];
    }
}

// ---------------------------------------------------------------------------
// Kernel 2: per-(b,n,h) score dots: s1,s2 (from Wh1) and o1,o2 (from Wh2)
// ---------------------------------------------------------------------------
__global__ __launch_bounds__(256) void scores_kernel(
    const float* __restrict__ wh1, const float* __restrict__ wh2,
    const float* __restrict__ Ai, const float* __restrict__ Aj,
    float* __restrict__ s1, float* __restrict__ s2,
    float* __restrict__ o1, float* __restrict__ o2)
{
    int idx = blockIdx.x * blockDim.x + threadIdx.x;   // over B*N*H
    if (idx >= Bc * Nn * Hh) return;
    const int h = idx % Hh;
    const int n = (idx / Hh) % Nn;
    const int b = idx / (Hh * Nn);

    const float* r1 = wh1 + ((long)(b * Nn + n)) * Dd + h * DH;
    const float* r2 = wh2 + ((long)(b * Nn + n)) * Dd + h * DH;
    const float* ai = Ai + h * DH;
    const float* aj = Aj + h * DH;

    float a1 = 0.f, a2 = 0.f, a3 = 0.f, a4 = 0.f;
#pragma unroll 8
    for (int d = 0; d < DH; ++d) {
        const float w1v = r1[d], w2v = r2[d], aiv = ai[d], ajv = aj[d];
        a1 += w1v * aiv;  a2 += w1v * ajv;
        a3 += w2v * aiv;  a4 += w2v * ajv;
    }
    const int o = (b * Hh + h) * Nn + n;
    s1[o] = a1; s2[o] = a2; o1[o] = a3; o2[o] = a4;
}

// ---------------------------------------------------------------------------
// Kernel 3: exact 1.5-entmax per row (sort algorithm), one block per (b,h,i)
// ---------------------------------------------------------------------------
__global__ __launch_bounds__(256) void entmax_kernel(
    const float* __restrict__ s1, const float* __restrict__ s2,
    const float* __restrict__ o1, const float* __restrict__ o2,
    const int* __restrict__ adj, const int* __restrict__ nag,
    float* __restrict__ att)
{
    __shared__ float xs[Nn];
    __shared__ float srt[Nn];
    __shared__ float tau[Nn];
    __shared__ float redA[256];
    __shared__ float redB[256];
    __shared__ int   redI[256];

    const int tid = threadIdx.x;
    const int bid = blockIdx.x;
    const int i   = bid & (Nn - 1);
    const int h   = (bid >> 10) & (Hh - 1);
    const int b   = bid >> 13;
    const int na  = nag[0];

    const int  rowbase = (b * Hh + h) * Nn;
    const float s1i = s1[rowbase + i];
    const float o1i = o1[rowbase + i];
    const int   im  = i % na;
    const long  adjbase = ((long)b * Nn + i) * Nn;   // adj is [B,1,N,N]

    // build x = e/2, track max
    float lmax = -3.4e38f;
    for (int j = tid; j < Nn; j += 256) {
        float e = (im == (j % na)) ? (s1i + s2[rowbase + j])
                                   : (o1i + o2[rowbase + j]);
        if (adj[adjbase + j] == 0) e = -1.0e12f;
        const float x = 0.5f * e;
        xs[j] = x;
        lmax  = fmaxf(lmax, x);
    }
    redA[tid] = lmax;
    __syncthreads();
    for (int s = 128; s > 0; s >>= 1) {
        if (tid < s) redA[tid] = fmaxf(redA[tid], redA[tid + s]);
        __syncthreads();
    }
    const float xmax = redA[0];
    __syncthreads();
    for (int j = tid; j < Nn; j += 256) {
        const float x = xs[j] - xmax;
        xs[j]  = x;
        srt[j] = x;
    }
    __syncthreads();

    // bitonic sort, descending (disjoint compare-exchange pairs per step)
    for (int k = 2; k <= Nn; k <<= 1) {
        for (int jj = k >> 1; jj > 0; jj >>= 1) {
            for (int idx = tid; idx < Nn; idx += 256) {
                const int ixj = idx ^ jj;
                if (ixj > idx) {
                    const float a = srt[idx];
                    const float c = srt[ixj];
                    const bool desc = ((idx & k) == 0);
                    if (desc ? (a < c) : (a > c)) { srt[idx] = c; srt[ixj] = a; }
                }
            }
            __syncthreads();
        }
    }

    // inclusive scans of srt and srt^2: chunk-of-4 sequential + block scan
    const int base = tid * 4;
    const float v0 = srt[base], v1 = srt[base + 1];
    const float v2 = srt[base + 2], v3 = srt[base + 3];
    const float c1 = v0 + v1, c2 = c1 + v2, c3 = c2 + v3;
    const float q0 = v0 * v0, q1 = v1 * v1, q2 = v2 * v2, q3 = v3 * v3;
    const float d1 = q0 + q1, d2 = d1 + q2, d3 = d2 + q3;
    redA[tid] = c3;
    redB[tid] = d3;
    __syncthreads();
    for (int off = 1; off < 256; off <<= 1) {
        const float pa = (tid >= off) ? redA[tid - off] : 0.0f;
        const float pb = (tid >= off) ? redB[tid - off] : 0.0f;
        __syncthreads();
        redA[tid] += pa;
        redB[tid] += pb;
        __syncthreads();
    }
    const float exS = redA[tid] - c3;   // exclusive prefix (sum)
    const float exQ = redB[tid] - d3;   // exclusive prefix (sum of squares)

    const float sv[4] = { v0, v1, v2, v3 };
    const float cs[4] = { exS + v0, exS + c1, exS + c2, exS + c3 };
    const float cq[4] = { exQ + q0, exQ + d1, exQ + d2, exQ + d3 };
    int cnt = 0;
#pragma unroll
    for (int u = 0; u < 4; ++u) {
        const float rho   = (float)(base + u + 1);
        const float mean  = cs[u] / rho;
        const float msq   = cq[u] / rho;
        const float ssv   = rho * (msq - mean * mean);
        const float delta = (1.0f - ssv) / rho;
        const float t     = mean - ((delta > 0.0f) ? sqrtf(delta) : 0.0f);
        tau[base + u] = t;
        cnt += (t <= sv[u]) ? 1 : 0;
    }
    redI[tid] = cnt;
    __syncthreads();
    for (int s = 128; s > 0; s >>= 1) {
        if (tid < s) redI[tid] += redI[tid + s];
        __syncthreads();
    }
    const int   support = redI[0];
    const float tstar   = tau[support - 1];

    float* outrow = att + ((long)(b * Hh + h) * Nn + i) * Nn;
    for (int j = tid; j < Nn; j += 256) {
        float y = xs[j] - tstar;
        y = fmaxf(y, 0.0f);
        outrow[j] = y * y;
    }
}

// ---------------------------------------------------------------------------
// Kernel 4: h' = (att*mask)@Wh1t + (att*(1-mask))@Wh2t  via masked-A WMMA.
// 4 waves per block (64-row M-tile). Wh1/Wh2 K-chunks (128 rows x 32 cols)
// are cooperatively staged into LDS with coalesced float4 loads and shared
// by all 4 waves (4x L2-traffic reduction). LDS row stride padded to 40
// floats so the two half-waves (rows k and k+2) hit disjoint bank ranges.
// Self/other mask applied in registers on the A (att) tile; both WMMA
// chains accumulate into one C tile, so att is read exactly once.
// ---------------------------------------------------------------------------
static constexpr int KC  = 128;   // K-chunk rows staged in LDS
static constexpr int LDP = 40;    // padded LDS row stride (floats)

__global__ __launch_bounds__(128) void aggregate_kernel(
    const float* __restrict__ att, const float* __restrict__ wh1,
    const float* __restrict__ wh2, const int* __restrict__ nag,
    float* __restrict__ hp)
{
    __shared__ float ldsW1[KC][LDP];   // 20.5 KB
    __shared__ float ldsW2[KC][LDP];   // 20.5 KB

    const int tid  = threadIdx.x;
    const int wave = tid >> 5;           // 0..3
    const int lane = tid & 31;
    const int m    = lane & 15;
    const int kh   = (lane >> 4) << 1;   // 0 or 2
    const int bh   = blockIdx.x;
    const int b    = bh / Hh;
    const int h    = bh % Hh;
    const int i0   = blockIdx.y * 64 + wave * 16;
    const int na   = nag[0];

    const int row    = i0 + m;
    const int rowmod = row % na;
    const float* attrow = att + ((long)bh * Nn + row) * Nn;

    v8f accS0 = {};   // dh columns 0..15
    v8f accS1 = {};   // dh columns 16..31

    for (int kc = 0; kc < Nn; kc += KC) {
        __syncthreads();
        // cooperative stage: thread t loads row (kc+t), 32 cols, as 8 float4
        {
            const float* src1 = wh1 + ((long)(b * Nn + kc + tid)) * Dd + h * DH;
            const float* src2 = wh2 + ((long)(b * Nn + kc + tid)) * Dd + h * DH;
            float4* dst1 = (float4*)&ldsW1[tid][0];
            float4* dst2 = (float4*)&ldsW2[tid][0];
#pragma unroll
            for (int q = 0; q < 8; ++q) {
                dst1[q] = ((const float4*)src1)[q];
                dst2[q] = ((const float4*)src2)[q];
            }
        }
        __syncthreads();

        for (int k0 = 0; k0 < KC; k0 += 4) {
            const int j0 = kc + k0 + kh;       // global column (K) index
            const int j1 = j0 + 1;
            const v2f av = *(const v2f*)(attrow + j0);
            const bool self0 = ((j0 % na) == rowmod);
            const bool self1 = ((j1 % na) == rowmod);
            v2f aS, aO;
            aS.x = self0 ? av.x : 0.0f;  aS.y = self1 ? av.y : 0.0f;
            aO.x = self0 ? 0.0f : av.x;  aO.y = self1 ? 0.0f : av.y;

            const int l0 = k0 + kh;            // LDS row for K element kh
            v2f b1_0; b1_0.x = ldsW1[l0][m];      b1_0.y = ldsW1[l0 + 1][m];
            v2f b1_1; b1_1.x = ldsW1[l0][16 + m]; b1_1.y = ldsW1[l0 + 1][16 + m];
            v2f b2_0; b2_0.x = ldsW2[l0][m];      b2_0.y = ldsW2[l0 + 1][m];
            v2f b2_1; b2_1.x = ldsW2[l0][16 + m]; b2_1.y = ldsW2[l0 + 1][16 + m];

            accS0 = __builtin_amdgcn_wmma_f32_16x16x4_f32(false, aS, false, b1_0,
                                                          (short)0, accS0, false, false);
            accS0 = __builtin_amdgcn_wmma_f32_16x16x4_f32(false, aO, false, b2_0,
                                                          (short)0, accS0, false, false);
            accS1 = __builtin_amdgcn_wmma_f32_16x16x4_f32(false, aS, false, b1_1,
                                                          (short)0, accS1, false, false);
            accS1 = __builtin_amdgcn_wmma_f32_16x16x4_f32(false, aO, false, b2_1,
                                                          (short)0, accS1, false, false);
        }
    }

    const int rbase = i0 + ((lane >> 4) << 3);
#pragma unroll
    for (int r = 0; r < 8; ++r) {
        float* o = hp + ((long)(b * Nn + rbase + r)) * Dd + h * DH;
        o[m]      = accS0[r];
        o[16 + m] = accS1[r];
    }
}

// ---------------------------------------------------------------------------
// Kernel 5: out = elu(hp @ Wo^T + bo)   (fp32 WMMA 16x16x4 + epilogue)
// ---------------------------------------------------------------------------
__global__ __launch_bounds__(32) void outproj_kernel(
    const float* __restrict__ hp, const float* __restrict__ Wo,
    const float* __restrict__ bo, float* __restrict__ out)
{
    const int lane = threadIdx.x & 31;
    const int m    = lane & 15;
    const int kh   = (lane >> 4) << 1;
    const int col0 = blockIdx.x * 16;
    const int row0 = blockIdx.y * 16;

    v8f acc = {};
    const float* aRow = hp + (long)(row0 + m) * Dd;
    const float* bRow = Wo + (long)(col0 + m) * Dd;
    for (int k0 = 0; k0 < Dd; k0 += 4) {
        v2f a  = *(const v2f*)(aRow + k0 + kh);
        v2f bb = *(const v2f*)(bRow + k0 + kh);
        acc = __builtin_amdgcn_wmma_f32_16x16x4_f32(false, a, false, bb,
                                                    (short)0, acc, false, false);
    }

    const int rbase = row0 + ((lane >> 4) << 3);
    const int c     = col0 + m;
    const float bias = bo[c];
#pragma unroll
    for (int r = 0; r < 8; ++r) {
        float v = acc[r] + bias;
        v = (v > 0.0f) ? v : (expf(v) - 1.0f);   // elu
        out[(long)(rbase + r) * Dd + c] = v;
    }
}

// ---------------------------------------------------------------------------
// Host launcher
// ---------------------------------------------------------------------------
extern "C" void kernel_launch(void* const* d_in, const int* in_sizes, int n_in,
                              void* d_out, int out_size, void* d_ws, size_t ws_size,
                              hipStream_t stream) {
    (void)in_sizes; (void)n_in; (void)out_size; (void)ws_size;

    const float* hmat = (const float*)d_in[0];
    const int*   nag  = (const int*)  d_in[1];   // num_agent (device scalar)
    const int*   adj  = (const int*)  d_in[2];   // [B,1,N,N] int32
    const float* W1   = (const float*)d_in[3];
    const float* W2   = (const float*)d_in[4];
    const float* Ai   = (const float*)d_in[5];
    const float* Aj   = (const float*)d_in[6];
    const float* Wo   = (const float*)d_in[7];
    const float* bo   = (const float*)d_in[8];
    float*       out  = (float*)d_out;

    // workspace layout (floats)
    float* ws  = (float*)d_ws;
    const size_t whN = (size_t)Bc * Nn * Dd;     // 1,048,576
    const size_t scN = (size_t)Bc * Hh * Nn;     //    32,768
    float* wh1 = ws;
    float* wh2 = wh1 + whN;
    float* s1  = wh2 + whN;
    float* s2  = s1 + scN;
    float* o1  = s2 + scN;
    float* o2  = o1 + scN;
    float* hp  = o2 + scN;
    float* att = hp + whN;                       // B*H*N*N = 33,554,432 floats

    proj12_kernel<<<dim3(Dd / 16, (Bc * Nn) / 16), 32, 0, stream>>>(
        hmat, W1, W2, wh1, wh2);

    scores_kernel<<<(Bc * Nn * Hh + 255) / 256, 256, 0, stream>>>(
        wh1, wh2, Ai, Aj, s1, s2, o1, o2);

    entmax_kernel<<<Bc * Hh * Nn, 256, 0, stream>>>(
        s1, s2, o1, o2, adj, nag, att);

    aggregate_kernel<<<dim3(Bc * Hh, Nn / 64), 128, 0, stream>>>(
        att, wh1, wh2, nag, hp);

    outproj_kernel<<<dim3(Dd / 16, (Bc * Nn) / 16), 32, 0, stream>>>(
        hp, Wo, bo, out);
}